// WaveletTransform_53678501266080
// MI455X (gfx1250) — compile-verified
//
#include <hip/hip_runtime.h>
#include <hip/hip_bf16.h>
#include <stdint.h>

// ---------------------------------------------------------------------------
// Haar 2x2 wavelet (LL,LH,HL,HH), x: (8,64,512,512) f32 -> 4x (8,64,256,256)
//
// Pure HBM-streaming kernel (~1.07 GB traffic, ~46us roofline @ 23.3 TB/s).
// CDNA5 path: double-buffered GLOBAL_LOAD_ASYNC_TO_LDS_B128 (ASYNCcnt)
// pipeline + ds_load_b128 reads + non-temporal stores (no reuse, 1GB >> L2).
//
// Round-1 fix: the shared array must ESCAPE into the async-copy asm (dummy
// pointer operand), otherwise LLVM proves the asm can't write LDS, folds the
// LDS loads to undef, and the whole compute collapses to NaN stores.
// ---------------------------------------------------------------------------

typedef float v2f __attribute__((ext_vector_type(2)));
typedef float v4f __attribute__((ext_vector_type(4)));

#define TILE_FLOATS 2048u          // 4 input rows x 512 cols = 8 KB
#define TILE_BYTES  8192u
#define TILES_TOTAL 65536u         // (8*64*512*512) / 2048
#define NBLOCKS     2048
#define TILES_PER_BLOCK 32         // TILES_TOTAL / NBLOCKS
#define PLANE       33554432ull    // 512 * 256 * 256 elements per output plane

// Async copy 16B global -> LDS (CDNA5, ASYNCcnt-tracked).
// %0 (VDST) = LDS byte address in a VGPR, %1 (VADDR) = 64-bit global address.
// `lds_escape` is not referenced in the template; it exists so the compiler
// knows this asm may write the shared array (prevents undef-folding of the
// subsequent LDS reads).
__device__ __forceinline__ void async_copy_b128(uint32_t lds_byte_off,
                                                const float* gaddr,
                                                float* lds_escape) {
    asm volatile("global_load_async_to_lds_b128 %0, %1, off"
                 :: "v"(lds_byte_off), "v"((uint64_t)(uintptr_t)gaddr),
                    "v"(lds_escape)
                 : "memory");
}

__global__ __launch_bounds__(256) void
haar2x2_async_kernel(const float* __restrict__ x, float* __restrict__ out) {
    // Single shared allocation -> starts at LDS offset 0, so the raw byte
    // offsets used by the async-copy asm and the C++ reads agree.
    __shared__ float lds[2 * TILE_FLOATS];   // 16 KB double buffer

    const uint32_t tid      = threadIdx.x;
    const uint32_t tileBase = blockIdx.x * TILES_PER_BLOCK;

    // ---- prologue: stage tile 0 into buffer 0 (2 async b128 per thread) ----
    {
        const float* g = x + (size_t)tileBase * TILE_FLOATS;
        async_copy_b128(tid * 16u,         g + tid * 4u,          lds);
        async_copy_b128(tid * 16u + 4096u, g + 1024u + tid * 4u,  lds);
    }

    for (int j = 0; j < TILES_PER_BLOCK; ++j) {
        const uint32_t tile = tileBase + (uint32_t)j;
        const uint32_t buf  = (uint32_t)j & 1u;

        if (j + 1 < TILES_PER_BLOCK) {
            // Prefetch next tile into the other buffer. That buffer was last
            // read at iteration j-1 and is protected by the trailing barrier.
            const float*   g   = x + (size_t)(tile + 1u) * TILE_FLOATS;
            const uint32_t dst = (((uint32_t)j + 1u) & 1u) * TILE_BYTES;
            async_copy_b128(dst + tid * 16u,         g + tid * 4u,         lds);
            async_copy_b128(dst + tid * 16u + 4096u, g + 1024u + tid * 4u, lds);
            // Async loads retire in order: <=2 outstanding means this wave's
            // 2 loads for tile j have landed; only the prefetch is in flight.
            asm volatile("s_wait_asynccnt 0x2" ::: "memory");
        } else {
            asm volatile("s_wait_asynccnt 0x0" ::: "memory");
        }
        __syncthreads();   // tile j visible to all 8 waves

        // ---- compute: 2 output rows x 256 cols from 4 staged input rows ----
        const uint32_t o = tid >> 7;     // output row within tile (0..1)
        const uint32_t c = tid & 127u;   // column pair index (2 cols/thread)

        const uint32_t fbase = buf * TILE_FLOATS + o * 1024u + 4u * c;
        const v4f T  = *(const v4f*)&lds[fbase];         // input row 2o
        const v4f Bt = *(const v4f*)&lds[fbase + 512u];  // input row 2o+1

        const float s0 = T.x + T.y,  d0 = T.x - T.y;
        const float s1 = Bt.x + Bt.y, d1 = Bt.x - Bt.y;
        const float s2 = T.z + T.w,  d2 = T.z - T.w;
        const float s3 = Bt.z + Bt.w, d3 = Bt.z - Bt.w;

        v2f LL, LH, HL, HH;
        LL.x = (s0 + s1) * 0.5f;  LL.y = (s2 + s3) * 0.5f;
        LH.x = (s0 - s1) * 0.5f;  LH.y = (s2 - s3) * 0.5f;
        HL.x = (d0 + d1) * 0.5f;  HL.y = (d2 + d3) * 0.5f;
        HH.x = (d0 - d1) * 0.5f;  HH.y = (d2 - d3) * 0.5f;

        const uint32_t n = tile >> 7;      // image index (B*C)
        const uint32_t r = tile & 127u;    // output-row-pair within image
        const size_t obase = (size_t)n * 65536u
                           + (size_t)(2u * r + o) * 256u
                           + 2u * c;

        // Streaming (non-temporal) stores: no reuse, don't pollute L2.
        __builtin_nontemporal_store(LL, (v2f*)(out + obase));
        __builtin_nontemporal_store(LH, (v2f*)(out + PLANE + obase));
        __builtin_nontemporal_store(HL, (v2f*)(out + 2ull * PLANE + obase));
        __builtin_nontemporal_store(HH, (v2f*)(out + 3ull * PLANE + obase));

        __syncthreads();   // all reads of buf done before it is refilled
    }
}

extern "C" void kernel_launch(void* const* d_in, const int* in_sizes, int n_in,
                              void* d_out, int out_size, void* d_ws, size_t ws_size,
                              hipStream_t stream) {
    const float* x   = (const float*)d_in[0];
    float*       out = (float*)d_out;
    hipLaunchKernelGGL(haar2x2_async_kernel, dim3(NBLOCKS), dim3(256), 0, stream,
                       x, out);
}